// LinearHebbianLayer_20770461843594
// MI455X (gfx1250) — compile-verified
//
#include <hip/hip_runtime.h>
#include <math.h>

#define IN_F 4096
#define OUT_F 4096

typedef __attribute__((ext_vector_type(2))) float v2f;
typedef __attribute__((ext_vector_type(4))) float v4f;
typedef __attribute__((ext_vector_type(8))) float v8f;

// ---------------------------------------------------------------------------
// Kernel 1: post[n] = tanh( sum_k pre[k] * W[k][n] )  via V_WMMA_F32_16X16X4_F32
//
// Grid: OUT_F/16 = 256 blocks, 256 threads = 8 waves. Block owns a 16-column
// tile [n0, n0+16). Wave w covers k in [w*512, w*512+512), 4 k per WMMA.
// A (16x4 f32) = pre[k0..k0+3] broadcast on all 16 rows; B (4x16 f32) =
// W[k0..k0+3][n0..n0+15]. ISA layout: VGPR j holds K=j (lanes 0-15) and
// K=j+2 (lanes 16-31) for both A and B -> per-lane base row ka = k0+2*khalf.
// All 16 rows of D are identical; row 0 (acc[0], lanes 0-15) is the partial
// dot for column n0+lane. Cross-wave reduction through LDS, then tanh.
// ---------------------------------------------------------------------------
__global__ __launch_bounds__(256) void gemv_tanh_wmma(
    const float* __restrict__ pre, const float* __restrict__ W,
    float* __restrict__ post)
{
  const int lane  = threadIdx.x & 31;
  const int wave  = threadIdx.x >> 5;
  const int n0    = blockIdx.x << 4;
  const int col   = n0 + (lane & 15);
  const int khalf = lane >> 4;          // 0: rows k0,k0+1  1: rows k0+2,k0+3

  v8f acc = {};
  const int kbeg = wave * (IN_F / 8);
  const int kend = kbeg + (IN_F / 8);

  #pragma unroll 4
  for (int k0 = kbeg; k0 < kend; k0 += 4) {
    const int ka = k0 + 2 * khalf;
    v2f a, b;
    a.x = pre[ka];
    a.y = pre[ka + 1];
    const float* wp = W + (size_t)ka * OUT_F + col;
    b.x = wp[0];          // row ka,   column col
    b.y = wp[OUT_F];      // row ka+1, column col
    // 8 args: (neg_a, A, neg_b, B, c_mod, C, reuse_a, reuse_b)
    acc = __builtin_amdgcn_wmma_f32_16x16x4_f32(
        false, a, false, b, (short)0, acc, false, false);
  }

  __shared__ float red[8][16];
  if (lane < 16) red[wave][lane] = acc[0];   // D row 0, N = lane
  __syncthreads();
  if (threadIdx.x < 16) {
    float s = 0.f;
    #pragma unroll
    for (int w = 0; w < 8; ++w) s += red[w][threadIdx.x];
    post[n0 + threadIdx.x] = tanhf(s);
  }
}

// ---------------------------------------------------------------------------
// Kernel 2: w_new = W + eta*(A*(pre_col*post_row) + B*pre_col + C*post_row + D)
//
// 4 consecutive elements per thread: the 5-float-per-element h stream becomes
// five 16B-aligned v4f loads (80 bytes, global_load_b128). h is 320 MB and
// read once -> non-temporal loads; w_new is write-once -> non-temporal store.
// W stays regular-temporal so the 192 MB L2 can retain it from kernel 1.
// ---------------------------------------------------------------------------
__global__ __launch_bounds__(256) void hebb_update(
    const float* __restrict__ pre, const float* __restrict__ W,
    const float* __restrict__ h, const float* __restrict__ post,
    float* __restrict__ Wout)
{
  const size_t e = ((size_t)blockIdx.x * blockDim.x + threadIdx.x) * 4;
  const int i = (int)(e >> 12);            // row   (e / OUT_F)
  const int o = (int)(e & (OUT_F - 1));    // col   (e % OUT_F), multiple of 4

  const float p  = pre[i];
  const v4f  pv  = *(const v4f*)(post + o);
  const v4f  wv  = *(const v4f*)(W + e);

  const float* hb = h + e * 5;             // byte offset e*20, 16B aligned
  const v4f h0 = __builtin_nontemporal_load((v4f*)(hb + 0));
  const v4f h1 = __builtin_nontemporal_load((v4f*)(hb + 4));
  const v4f h2 = __builtin_nontemporal_load((v4f*)(hb + 8));
  const v4f h3 = __builtin_nontemporal_load((v4f*)(hb + 12));
  const v4f h4 = __builtin_nontemporal_load((v4f*)(hb + 16));

  // element j fields: [eta, A, B, C, D] at hb[5j .. 5j+4]
  const float eta0 = h0.x, A0 = h0.y, B0 = h0.z, C0 = h0.w, D0 = h1.x;
  const float eta1 = h1.y, A1 = h1.z, B1 = h1.w, C1 = h2.x, D1 = h2.y;
  const float eta2 = h2.z, A2 = h2.w, B2 = h3.x, C2 = h3.y, D2 = h3.z;
  const float eta3 = h3.w, A3 = h4.x, B3 = h4.y, C3 = h4.z, D3 = h4.w;

  v4f r;
  r.x = wv.x + eta0 * (A0 * (p * pv.x) + B0 * p + C0 * pv.x + D0);
  r.y = wv.y + eta1 * (A1 * (p * pv.y) + B1 * p + C1 * pv.y + D1);
  r.z = wv.z + eta2 * (A2 * (p * pv.z) + B2 * p + C2 * pv.z + D2);
  r.w = wv.w + eta3 * (A3 * (p * pv.w) + B3 * p + C3 * pv.w + D3);

  __builtin_nontemporal_store(r, (v4f*)(Wout + e));
}

extern "C" void kernel_launch(void* const* d_in, const int* in_sizes, int n_in,
                              void* d_out, int out_size, void* d_ws, size_t ws_size,
                              hipStream_t stream) {
  const float* pre    = (const float*)d_in[0];
  const float* weight = (const float*)d_in[1];
  const float* h      = (const float*)d_in[2];

  float* post = (float*)d_out;        // first OUT_F floats of output
  float* wout = post + OUT_F;         // then IN_F*OUT_F floats (w_new)

  gemv_tanh_wmma<<<OUT_F / 16, 256, 0, stream>>>(pre, weight, post);

  const int n_chunks = (IN_F * OUT_F) / 4;          // 4 elements per thread
  hebb_update<<<n_chunks / 256, 256, 0, stream>>>(pre, weight, h, post, wout);
}